// GCNActor_87797721465076
// MI455X (gfx1250) — compile-verified
//
#include <hip/hip_runtime.h>
#include <math.h>

typedef __attribute__((ext_vector_type(2))) float v2f;
typedef __attribute__((ext_vector_type(8))) float v8f;

#define HID 128  // hidden width (K is always 128 for every GEMM here)

// ---------------- degree / normalization ----------------

__global__ void k_deg_init(float* deg, int n) {
  int i = blockIdx.x * blockDim.x + threadIdx.x;
  if (i < n) deg[i] = 1.0f;  // self-loop contributes 1 to every node's degree
}

__global__ void k_deg_scatter(const int* __restrict__ dst, float* __restrict__ deg, int e) {
  int i = blockIdx.x * blockDim.x + threadIdx.x;
  if (i < e) atomicAdd(&deg[dst[i]], 1.0f);
}

__global__ void k_dinv(float* deg, int n) {
  int i = blockIdx.x * blockDim.x + threadIdx.x;
  if (i < n) {
    float d = deg[i];
    deg[i] = (d > 0.0f) ? rsqrtf(d) : 0.0f;  // in place: buffer becomes dinv
  }
}

// ---------------- WMMA GEMM: OUT = epilogue( act(X) @ W ) ----------------
// X: [nRows,128] row-major, W: [128,COUT] row-major.
// One block computes a 16-row panel; wave w owns columns [16w,16w+16).
// RELU_IN: apply relu to X while staging into LDS.
// POST: 0 = raw, 1 = +bias, 2 = sigmoid(+bias)
// COUT: compile-time output width (128 or 64) -> shift/immediate addressing.
template <int RELU_IN, int POST, int COUT>
__global__ void k_gemm_wmma(const float* __restrict__ X, const float* __restrict__ W,
                            const float* __restrict__ bias, float* __restrict__ OUT,
                            int nRows) {
  // pad 128->132: LDS read pattern (lane m reads As[m][k]) hits 32 distinct banks
  __shared__ float As[16][132];

  const int r0   = blockIdx.x * 16;
  const int tid  = threadIdx.x;
  const int nthr = blockDim.x;

  // stage the 16x128 A panel, coalesced, optional ReLU
  for (int i = tid; i < 16 * HID; i += nthr) {
    int row = i >> 7, k = i & 127;
    float v = 0.0f;
    if (r0 + row < nRows) v = X[(r0 + row) * HID + k];
    if (RELU_IN) v = fmaxf(v, 0.0f);
    As[row][k] = v;
  }
  __syncthreads();

  const int wave = tid >> 5;
  const int lane = tid & 31;
  const int hlf  = lane >> 4;   // 0: lanes 0-15, 1: lanes 16-31
  const int m    = lane & 15;
  const int col0 = wave * 16;

  v8f c = {};
  for (int k0 = 0; k0 < HID; k0 += 4) {
    int ka = k0 + 2 * hlf;  // this half-wave covers K = ka, ka+1
    v2f a, b;
    // A 16x4 fragment (ISA 7.12.2): lane m holds row m, vgpr0=K(2*hlf), vgpr1=K(2*hlf+1)
    a.x = As[m][ka];
    a.y = As[m][ka + 1];
    // B 4x16 fragment (mirrored): lane m holds col m of the same K pair
    b.x = W[ka * COUT + col0 + m];
    b.y = W[(ka + 1) * COUT + col0 + m];
    c = __builtin_amdgcn_wmma_f32_16x16x4_f32(false, a, false, b, (short)0, c,
                                              false, false);
  }

  // epilogue value transform
  float bv = (POST >= 1) ? bias[col0 + m] : 0.0f;
  float vals[8];
#pragma unroll
  for (int r = 0; r < 8; ++r) {
    float v = c[r];
    if (POST >= 1) v += bv;
    if (POST == 2) v = __builtin_amdgcn_rcpf(1.0f + __expf(-v));  // fast sigmoid
    vals[r] = v;
  }

  // C/D layout: vgpr r -> row r + 8*hlf, col = lane%16.
  // Base pointer per lane; r*COUT becomes an immediate offset in the unroll.
  float* outp = OUT + (r0 + 8 * hlf) * COUT + col0 + m;
  if (r0 + 16 <= nRows) {
    // fast path: whole panel in-bounds (always true when nRows % 16 == 0)
#pragma unroll
    for (int r = 0; r < 8; ++r) outp[r * COUT] = vals[r];
  } else {
#pragma unroll
    for (int r = 0; r < 8; ++r)
      if (r0 + 8 * hlf + r < nRows) outp[r * COUT] = vals[r];
  }
}

// ---------------- ACC = bias + T * dinv^2 (self-loop message) ----------------
__global__ void k_init_acc(float* __restrict__ ACC, const float* __restrict__ T,
                           const float* __restrict__ bias,
                           const float* __restrict__ dinv, int total) {
  int i = blockIdx.x * blockDim.x + threadIdx.x;
  if (i >= total) return;
  int node = i >> 7;       // /128
  int col  = i & 127;
  float dv = dinv[node];
  ACC[i] = bias[col] + T[i] * dv * dv;
}

// ---------------- edge scatter: one wave per edge, float4 row slice ----------------
__global__ void k_edge_scatter(const float* __restrict__ T, float* __restrict__ ACC,
                               const int* __restrict__ src, const int* __restrict__ dst,
                               const float* __restrict__ dinv, int e) {
  int g    = blockIdx.x * blockDim.x + threadIdx.x;
  int lane = g & 31;
  // wave-uniform edge id as SGPR -> scalar loads for indices / norms
  int wid  = __builtin_amdgcn_readfirstlane(g >> 5);
  if (wid >= e) return;
  int s = src[wid], d = dst[wid];
  float w = dinv[s] * dinv[d];
  // lane covers columns [4*lane, 4*lane+4): one coalesced 512B row per wave
  float4 v = ((const float4*)(T + (size_t)s * HID))[lane];
  float* out = ACC + (size_t)d * HID + lane * 4;
  atomicAdd(out + 0, v.x * w);
  atomicAdd(out + 1, v.y * w);
  atomicAdd(out + 2, v.z * w);
  atomicAdd(out + 3, v.w * w);
}

// ---------------- host orchestration ----------------
extern "C" void kernel_launch(void* const* d_in, const int* in_sizes, int n_in,
                              void* d_out, int out_size, void* d_ws, size_t ws_size,
                              hipStream_t stream) {
  const float* x   = (const float*)d_in[0];
  const int*   ei  = (const int*)d_in[1];   // [2,E] flat: first E = src, next E = dst
  const float* W1  = (const float*)d_in[2];
  const float* b1  = (const float*)d_in[3];
  const float* W2  = (const float*)d_in[4];
  const float* b2  = (const float*)d_in[5];
  const float* W3  = (const float*)d_in[6];
  const float* b3  = (const float*)d_in[7];
  const float* Wd1 = (const float*)d_in[8];
  const float* bd1 = (const float*)d_in[9];
  const float* Wd2 = (const float*)d_in[10];
  const float* bd2 = (const float*)d_in[11];

  const int N = in_sizes[0] / HID;   // 50000
  const int E = in_sizes[1] / 2;     // 800000
  const int* srcI = ei;
  const int* dstI = ei + E;

  // workspace: dinv[N] | T[N*128] | ACC[N*128]  (~51.6 MB)
  float* dinv = (float*)d_ws;
  float* T    = dinv + N;
  float* ACC  = T + (size_t)N * HID;

  const int total = N * HID;
  const int gN    = (N + 255) / 256;
  const int gE    = (E + 255) / 256;
  const int gTot  = (total + 255) / 256;
  const int gEdge = (int)(((size_t)E * 32 + 255) / 256);
  const int gRow  = (N + 15) / 16;

  // normalization
  k_deg_init<<<gN, 256, 0, stream>>>(dinv, N);
  k_deg_scatter<<<gE, 256, 0, stream>>>(dstI, dinv, E);
  k_dinv<<<gN, 256, 0, stream>>>(dinv, N);

  // GCN layer 1 (input x, no relu on input)
  k_gemm_wmma<0, 0, HID><<<gRow, 256, 0, stream>>>(x, W1, nullptr, T, N);
  k_init_acc<<<gTot, 256, 0, stream>>>(ACC, T, b1, dinv, total);
  k_edge_scatter<<<gEdge, 256, 0, stream>>>(T, ACC, srcI, dstI, dinv, E);

  // GCN layer 2 (relu of previous ACC folded into A-load)
  k_gemm_wmma<1, 0, HID><<<gRow, 256, 0, stream>>>(ACC, W2, nullptr, T, N);
  k_init_acc<<<gTot, 256, 0, stream>>>(ACC, T, b2, dinv, total);
  k_edge_scatter<<<gEdge, 256, 0, stream>>>(T, ACC, srcI, dstI, dinv, E);

  // GCN layer 3
  k_gemm_wmma<1, 0, HID><<<gRow, 256, 0, stream>>>(ACC, W3, nullptr, T, N);
  k_init_acc<<<gTot, 256, 0, stream>>>(ACC, T, b3, dinv, total);
  k_edge_scatter<<<gEdge, 256, 0, stream>>>(T, ACC, srcI, dstI, dinv, E);

  // dense head: T = relu(ACC)@Wd1 + bd1 (relu of T folded into next GEMM)
  k_gemm_wmma<1, 1, HID><<<gRow, 256, 0, stream>>>(ACC, Wd1, bd1, T, N);
  // out = sigmoid(relu(T)@Wd2 + bd2), Cout=64 -> 4 waves (128 threads)
  k_gemm_wmma<1, 2, 64><<<gRow, 128, 0, stream>>>(T, Wd2, bd2, (float*)d_out, N);
}